// DGLHyperGCNIIConv_27831388078180
// MI455X (gfx1250) — compile-verified
//
#include <hip/hip_runtime.h>

typedef float v2f __attribute__((ext_vector_type(2)));
typedef float v8f __attribute__((ext_vector_type(8)));

#define DIM 128
#define WPAD 130   // padded LDS row stride (floats) to avoid 64-bank conflicts

// ---------------------------------------------------------------------------
// Scatter pass: for each nnz, one wave (32 lanes x float4) gathers a 128-float
// row from src_rows[src_idx[e]] (optionally scaled by src_scale[src]) and
// atomically accumulates into outrows[dst_idx[e]].
// ---------------------------------------------------------------------------
__global__ __launch_bounds__(256) void hyper_scatter_sum(
    const float* __restrict__ src_rows, const int* __restrict__ src_idx,
    const int* __restrict__ dst_idx, const float* __restrict__ src_scale,
    float* __restrict__ outrows, int nnz)
{
    long long t = (long long)blockIdx.x * blockDim.x + threadIdx.x;
    int e    = (int)(t >> 5);
    int lane = (int)(t & 31);
    if (e >= nnz) return;

    int s = src_idx[e];
    int d = dst_idx[e];
    float sc = src_scale ? src_scale[s] : 1.0f;

    const float4 v = *(const float4*)(src_rows + (size_t)s * DIM + lane * 4);
    float* o = outrows + (size_t)d * DIM + lane * 4;
    unsafeAtomicAdd(o + 0, v.x * sc);
    unsafeAtomicAdd(o + 1, v.y * sc);
    unsafeAtomicAdd(o + 2, v.z * sc);
    unsafeAtomicAdd(o + 3, v.w * sc);
}

// ---------------------------------------------------------------------------
// Fused epilogue: Xi = (1-a)*(Xv*degV) + a*X0 built in LDS per-wave (16 rows),
// then out = (1-b)*Xi + b*(Xi @ W^T) via V_WMMA_F32_16X16X4_F32.
// Block = 256 threads = 8 waves; each wave computes 16 rows x 128 cols.
// W (128x128 f32, 64KB) staged once per block in LDS, rows padded to 130.
// ---------------------------------------------------------------------------
__global__ __launch_bounds__(256) void hyper_final(
    const float* __restrict__ Xv, const float* __restrict__ X0,
    const float* __restrict__ degV, const float* __restrict__ alpha,
    const float* __restrict__ beta, const float* __restrict__ W,
    float* __restrict__ out, int n_nodes)
{
    __shared__ float sW [DIM * WPAD];        // W row-major, padded
    __shared__ float sXi[8 * 16 * WPAD];     // per-wave 16x128 Xi slab, padded

    const int t    = threadIdx.x;
    const int wave = t >> 5;
    const int lane = t & 31;

    const float a  = alpha[0];
    const float b  = beta[0];
    const float ia = 1.0f - a;
    const float ib = 1.0f - b;

    // --- stage W into LDS (float4 per thread-iteration) ---
    for (int i = t; i < DIM * (DIM / 4); i += 256) {
        int row = i >> 5;            // / (DIM/4)
        int c4  = (i & 31) << 2;     // column of the float4
        const float4 w4 = *(const float4*)(W + row * DIM + c4);
        float* p = &sW[row * WPAD + c4];
        p[0] = w4.x; p[1] = w4.y; p[2] = w4.z; p[3] = w4.w;
    }

    // --- build this wave's Xi slab in LDS ---
    const int rowbase = (blockIdx.x * 8 + wave) * 16;
    float* myXi = &sXi[wave * 16 * WPAD];
    for (int r = 0; r < 16; ++r) {
        int row = rowbase + r;
        int c   = lane * 4;
        float4 v = make_float4(0.f, 0.f, 0.f, 0.f);
        if (row < n_nodes) {
            float dv = degV[row];
            const float4 xv = *(const float4*)(Xv + (size_t)row * DIM + c);
            const float4 x0 = *(const float4*)(X0 + (size_t)row * DIM + c);
            v.x = ia * (xv.x * dv) + a * x0.x;
            v.y = ia * (xv.y * dv) + a * x0.y;
            v.z = ia * (xv.z * dv) + a * x0.z;
            v.w = ia * (xv.w * dv) + a * x0.w;
        }
        float* p = &myXi[r * WPAD + c];
        p[0] = v.x; p[1] = v.y; p[2] = v.z; p[3] = v.w;
    }

    __syncthreads();
    if (rowbase >= n_nodes) return;   // whole-wave uniform exit (EXEC stays all-1)

    // --- WMMA main loop: 8 N-tiles of 16, K=128 in steps of 4 ---
    v8f acc[8] = {};
    const int m     = lane & 15;           // M row (A) / N col (B,C)
    const int khalf = (lane >> 4) << 1;    // 0 or 2: which K-pair this lane holds

    for (int k0 = 0; k0 < DIM; k0 += 4) {
        // A frag: A[m][k0+khalf], A[m][k0+khalf+1]   (16x4 f32 layout)
        v2f afrag = *(const v2f*)&myXi[m * WPAD + k0 + khalf];
#pragma unroll
        for (int j = 0; j < 8; ++j) {
            // B frag: B[k][n] = W[n][k], n = j*16 + m  (4x16 f32 layout)
            v2f bfrag = *(const v2f*)&sW[(j * 16 + m) * WPAD + k0 + khalf];
            acc[j] = __builtin_amdgcn_wmma_f32_16x16x4_f32(
                false, afrag, false, bfrag, (short)0, acc[j], false, false);
        }
    }

    // --- epilogue: out = (1-b)*Xi + b*acc ; C/D layout per ISA 7.12.2 ---
    const int mhigh = (lane >> 4) << 3;    // lanes 16-31 hold rows M=8..15
#pragma unroll
    for (int j = 0; j < 8; ++j) {
        const int n = j * 16 + m;
#pragma unroll
        for (int r = 0; r < 8; ++r) {
            int ml  = r + mhigh;
            int row = rowbase + ml;
            if (row < n_nodes) {
                float xi = myXi[ml * WPAD + n];
                out[(size_t)row * DIM + n] = ib * xi + b * acc[j][r];
            }
        }
    }
}

// ---------------------------------------------------------------------------
// Launch: memset workspace -> scatter1 -> scatter2(+degE) -> fused WMMA finish
// ---------------------------------------------------------------------------
extern "C" void kernel_launch(void* const* d_in, const int* in_sizes, int n_in,
                              void* d_out, int out_size, void* d_ws, size_t ws_size,
                              hipStream_t stream) {
    const float* X     = (const float*)d_in[0];
    const float* X0    = (const float*)d_in[1];
    const float* degE  = (const float*)d_in[2];
    const float* degV  = (const float*)d_in[3];
    const float* alpha = (const float*)d_in[4];
    const float* beta  = (const float*)d_in[5];
    const float* W     = (const float*)d_in[6];
    const int*   g1s   = (const int*)d_in[7];
    const int*   g1d   = (const int*)d_in[8];
    const int*   g2s   = (const int*)d_in[9];
    const int*   g2d   = (const int*)d_in[10];

    const int n_edges = in_sizes[2];
    const int n_nodes = in_sizes[3];
    const int nnz     = in_sizes[7];

    float* Xe = (float*)d_ws;                          // [E, 128]
    float* Xvb = Xe + (size_t)n_edges * DIM;           // [N, 128]

    hipMemsetAsync(d_ws, 0,
                   ((size_t)n_edges + (size_t)n_nodes) * DIM * sizeof(float),
                   stream);

    // pass 1: nodes -> hyperedges (unscaled; degE folded into pass 2's gather)
    {
        long long threads = (long long)nnz * 32;
        int blocks = (int)((threads + 255) / 256);
        hyper_scatter_sum<<<blocks, 256, 0, stream>>>(X, g1s, g1d, nullptr, Xe, nnz);
    }
    // pass 2: (Xe * degE) -> nodes
    {
        long long threads = (long long)nnz * 32;
        int blocks = (int)((threads + 255) / 256);
        hyper_scatter_sum<<<blocks, 256, 0, stream>>>(Xe, g2s, g2d, degE, Xvb, nnz);
    }
    // fused: Xi build + f32 WMMA GEMM + residual mix
    {
        int blocks = (n_nodes + 127) / 128;   // 8 waves x 16 rows per block
        hyper_final<<<blocks, 256, 0, stream>>>(Xvb, X0, degV, alpha, beta, W,
                                                (float*)d_out, n_nodes);
    }
}